// WDNodeMPNNLayer_57681410785500
// MI455X (gfx1250) — compile-verified
//
#include <hip/hip_runtime.h>
#include <hip/hip_bf16.h>

typedef float v2f __attribute__((ext_vector_type(2)));
typedef float v4f __attribute__((ext_vector_type(4)));
typedef float v8f __attribute__((ext_vector_type(8)));

#define ND   128     // node_attr_dim
#define ED   16      // edge_attr_dim
#define HID  300     // hidden_dim
#define HP   320     // hidden padded to multiple of 64
#define K1   144     // ND + ED
#define NCG  5       // 64-wide column groups (HP/64)

__device__ __forceinline__ void atomAddF(float* p, float v) {
    // lowers to GLOBAL_ATOMIC_ADD_F32 on gfx1250
    __hip_atomic_fetch_add(p, v, __ATOMIC_RELAXED, __HIP_MEMORY_SCOPE_AGENT);
}

__device__ __forceinline__ v8f wmma4(v2f a, v2f b, v8f c) {
    return __builtin_amdgcn_wmma_f32_16x16x4_f32(false, a, false, b,
                                                 (short)0, c, false, false);
}

// B-panel LDS layout: lds[(k>>1)*128 + c*2 + (k&1)] = B[k][colBase+c]
// -> (B[k][c], B[k+1][c]) contiguous => one aligned ds_load_b64 per fragment.
__device__ __forceinline__ v2f bfrag(const float* ldsB, int pairRow, int col) {
    return *(const v2f*)(ldsB + pairRow * 128 + col * 2);
}

// ---------------- zero scratch (float4) ----------------
__global__ void k_zero4(float* __restrict__ p, long n4) {
    long i = (long)blockIdx.x * blockDim.x + threadIdx.x;
    if (i < n4) ((v4f*)p)[i] = (v4f){0.f, 0.f, 0.f, 0.f};
}

// ---------------- pad W0[144,300]->W0p[144,320], W[300,300]->Wp[300,320] ----
__global__ void k_prep_w(const float* __restrict__ W0, const float* __restrict__ W,
                         float* __restrict__ W0p, float* __restrict__ Wp) {
    int g = blockIdx.x * blockDim.x + threadIdx.x;
    if (g < K1 * HP) {
        int r = g / HP, c = g % HP;
        W0p[g] = (c < HID) ? W0[r * HID + c] : 0.f;
    } else {
        g -= K1 * HP;
        if (g < HID * HP) {
            int r = g / HP, c = g % HP;
            Wp[g] = (c < HID) ? W[r * HID + c] : 0.f;
        }
    }
}

// ---------------- edge scatter: inc[dst] += w*edge_attr, deg[dst] += 1 ------
__global__ void k_edge_inc_deg(const float* __restrict__ ea, const float* __restrict__ ew,
                               const int* __restrict__ ei, float* __restrict__ inc,
                               float* __restrict__ deg, int E) {
    int g = blockIdx.x * blockDim.x + threadIdx.x;
    if (g >= E * ED) return;
    int e = g >> 4, j = g & 15;
    int d = ei[E + e];
    float w = ew[e];
    atomAddF(&inc[(long)d * ED + j], w * ea[g]);
    if (j == 0) atomAddF(&deg[d], 1.0f);
}

// ---------------- edge message: agg[dst] += w * h0[src] (wave per edge) -----
__global__ void k_edge_msg(const float* __restrict__ h0, const float* __restrict__ ew,
                           const int* __restrict__ ei, float* __restrict__ agg, int E) {
    int wave = (int)((blockIdx.x * blockDim.x + threadIdx.x) >> 5);
    int lane = threadIdx.x & 31;
    if (wave >= E) return;
    int s = ei[wave], d = ei[E + wave];
    float w = ew[wave];
    const v4f* hrow = (const v4f*)(h0 + (long)s * HID);   // rows are 16B aligned
    float*     arow = agg + (long)d * HID;
    for (int q = lane; q < HID / 4; q += 32) {            // 75 float4 per edge
        v4f v = hrow[q];
        int j = q * 4;
        atomAddF(&arow[j + 0], w * v.x);
        atomAddF(&arow[j + 1], w * v.y);
        atomAddF(&arow[j + 2], w * v.z);
        atomAddF(&arow[j + 3], w * v.w);
    }
}

// ---------------- GEMM1: h0 = relu([x | inc] @ W0p + b0) --------------------
// block = 8 waves = 8 M-tiles sharing one LDS B-panel (K1 x 64, pair-interleaved);
// each wave computes a 16x64 strip = 4 WMMA accumulators per A fragment.
__global__ void k_gemm1_wmma(const float* __restrict__ x, const float* __restrict__ inc,
                             const float* __restrict__ W0p, const float* __restrict__ b0,
                             float* __restrict__ h0, int n, int mTiles) {
    extern __shared__ float ldsB[];                        // K1*64 floats
    int mtg = blockIdx.x / NCG, ncg = blockIdx.x % NCG;
    int tid = threadIdx.x;
    int colBase = ncg * 64;

    // stage B panel, pair-interleaved
    for (int g = tid; g < K1 * 64; g += 256) {
        int r = g >> 6, c = g & 63;
        ldsB[(r >> 1) * 128 + c * 2 + (r & 1)] =
            W0p[(long)r * HP + colBase + c];
    }
    __syncthreads();

    int waveId = tid >> 5;
    int mt = mtg * 8 + waveId; if (mt >= mTiles) mt = mTiles - 1;  // clamp (dup work ok)
    int lane = tid & 31, half = lane >> 4, l16 = lane & 15;
    int rowA = mt * 16 + l16; if (rowA >= n) rowA = n - 1;

    const float* xrow = x   + (long)rowA * ND;
    const float* irow = inc + (long)rowA * ED;

    v8f acc[4] = {};
    // K in [0,128): A from x
    for (int k = 0; k < ND; k += 4) {
        int ka = k + half * 2;                  // A lanes 0-15: K=k,k+1 ; 16-31: k+2,k+3
        v2f a = *(const v2f*)(xrow + ka);
        int p = (k >> 1) + half;                // matching B row-pair
        #pragma unroll
        for (int t = 0; t < 4; ++t)
            acc[t] = wmma4(a, bfrag(ldsB, p, t * 16 + l16), acc[t]);
    }
    // K in [128,144): A from inc
    for (int k = ND; k < K1; k += 4) {
        int ka = k + half * 2;
        v2f a = *(const v2f*)(irow + (ka - ND));
        int p = (k >> 1) + half;
        #pragma unroll
        for (int t = 0; t < 4; ++t)
            acc[t] = wmma4(a, bfrag(ldsB, p, t * 16 + l16), acc[t]);
    }

    int rbase = mt * 16 + half * 8;             // D VGPR r -> row r (+8 upper half)
    bool full = (mt * 16 + 16 <= n);            // wave-uniform
    #pragma unroll
    for (int t = 0; t < 4; ++t) {
        int colB = colBase + t * 16 + l16;
        if (colB < HID) {
            float bias = b0[colB];
            if (full) {
                #pragma unroll
                for (int r = 0; r < 8; ++r) {
                    float v = acc[t][r] + bias;
                    h0[(long)(rbase + r) * HID + colB] = v > 0.f ? v : 0.f;
                }
            } else {
                #pragma unroll
                for (int r = 0; r < 8; ++r) {
                    int row = rbase + r;
                    if (row < n) {
                        float v = acc[t][r] + bias;
                        h0[(long)row * HID + colB] = v > 0.f ? v : 0.f;
                    }
                }
            }
        }
    }
}

// ---------------- GEMM2: h = relu(h0 + (agg/deg) @ Wp + b) ------------------
__global__ void k_gemm2_wmma(const float* __restrict__ agg, const float* __restrict__ deg,
                             const float* __restrict__ Wp, const float* __restrict__ b,
                             const float* __restrict__ h0, float* __restrict__ h,
                             int n, int mTiles) {
    extern __shared__ float ldsB[];                        // HID*64 floats
    int mtg = blockIdx.x / NCG, ncg = blockIdx.x % NCG;
    int tid = threadIdx.x;
    int colBase = ncg * 64;

    for (int g = tid; g < HID * 64; g += 256) {
        int r = g >> 6, c = g & 63;
        ldsB[(r >> 1) * 128 + c * 2 + (r & 1)] =
            Wp[(long)r * HP + colBase + c];
    }
    __syncthreads();

    int waveId = tid >> 5;
    int mt = mtg * 8 + waveId; if (mt >= mTiles) mt = mTiles - 1;
    int lane = tid & 31, half = lane >> 4, l16 = lane & 15;
    int rowA = mt * 16 + l16; if (rowA >= n) rowA = n - 1;

    float dg = deg[rowA];
    float rdeg = 1.0f / (dg > 1.0f ? dg : 1.0f);   // scatter-mean folded into A load
    const float* arow = agg + (long)rowA * HID;

    v8f acc[4] = {};
    for (int k = 0; k < HID; k += 4) {             // 75 k-steps x 4 WMMAs
        int ka = k + half * 2;
        v2f a = *(const v2f*)(arow + ka);
        a.x *= rdeg; a.y *= rdeg;
        int p = (k >> 1) + half;
        #pragma unroll
        for (int t = 0; t < 4; ++t)
            acc[t] = wmma4(a, bfrag(ldsB, p, t * 16 + l16), acc[t]);
    }

    int rbase = mt * 16 + half * 8;
    bool full = (mt * 16 + 16 <= n);
    #pragma unroll
    for (int t = 0; t < 4; ++t) {
        int colB = colBase + t * 16 + l16;
        if (colB < HID) {
            float bias = b[colB];
            if (full) {
                #pragma unroll
                for (int r = 0; r < 8; ++r) {
                    long idx = (long)(rbase + r) * HID + colB;
                    float v = h0[idx] + acc[t][r] + bias;
                    h[idx] = v > 0.f ? v : 0.f;
                }
            } else {
                #pragma unroll
                for (int r = 0; r < 8; ++r) {
                    int row = rbase + r;
                    if (row < n) {
                        long idx = (long)row * HID + colB;
                        float v = h0[idx] + acc[t][r] + bias;
                        h[idx] = v > 0.f ? v : 0.f;
                    }
                }
            }
        }
    }
}

extern "C" void kernel_launch(void* const* d_in, const int* in_sizes, int n_in,
                              void* d_out, int out_size, void* d_ws, size_t ws_size,
                              hipStream_t stream) {
    (void)n_in; (void)out_size; (void)ws_size;
    const float* x  = (const float*)d_in[0];
    const float* ea = (const float*)d_in[1];
    const float* ew = (const float*)d_in[2];
    const float* W0 = (const float*)d_in[4];
    const float* b0 = (const float*)d_in[5];
    const float* W  = (const float*)d_in[6];
    const float* bb = (const float*)d_in[7];
    const int*   ei = (const int*)d_in[8];

    const int N = in_sizes[3];     // node_weight has N elements
    const int E = in_sizes[2];     // edge_weight has E elements

    // workspace layout (floats)
    float* ws  = (float*)d_ws;
    float* inc = ws;                                  // N*ED
    float* deg = inc + (long)N * ED;                  // N
    float* agg = deg + N;                             // N*HID
    float* W0p = agg + (long)N * HID;                 // K1*HP
    float* Wp  = W0p + (long)K1 * HP;                 // HID*HP

    float* h  = (float*)d_out;                        // output 0: h [N,HID]
    float* h0 = h + (long)N * HID;                    // output 1: h0 [N,HID]

    const int mTiles  = (N + 15) / 16;
    const int mGroups = (mTiles + 7) / 8;
    const int gemmBlocks = mGroups * NCG;

    // 1) zero inc|deg|agg (contiguous region; count divisible by 4)
    long nz4 = ((long)N * ED + N + (long)N * HID) / 4;
    k_zero4<<<(int)((nz4 + 255) / 256), 256, 0, stream>>>(ws, nz4);

    // 2) pad weights to 320 columns
    int prepTot = K1 * HP + HID * HP;
    k_prep_w<<<(prepTot + 255) / 256, 256, 0, stream>>>(W0, W, W0p, Wp);

    // 3) edge scatter -> inc, deg
    k_edge_inc_deg<<<(E * ED + 255) / 256, 256, 0, stream>>>(ea, ew, ei, inc, deg, E);

    // 4) GEMM1 -> h0 (fused concat + bias + relu), B panel in LDS
    k_gemm1_wmma<<<gemmBlocks, 256, K1 * 64 * sizeof(float), stream>>>(
        x, inc, W0p, b0, h0, N, mTiles);

    // 5) per-edge message scatter -> agg (float4 gather + f32 atomics)
    k_edge_msg<<<(int)(((long)E * 32 + 255) / 256), 256, 0, stream>>>(h0, ew, ei, agg, E);

    // 6) GEMM2 -> h (fused mean-normalize + residual + bias + relu)
    k_gemm2_wmma<<<gemmBlocks, 256, HID * 64 * sizeof(float), stream>>>(
        agg, deg, Wp, bb, h0, h, N, mTiles);
}